// Circuit_74895639707844
// MI455X (gfx1250) — compile-verified
//
#include <hip/hip_runtime.h>
#include <hip/hip_bf16.h>

// ---------------------------------------------------------------------------
// Coupled-mode ODE integrator for MI455X (gfx1250, wave32, WMMA).
//   Phase 1 (setup_kernel, 1 block): H -> U=expm(iH) -> M=U^T U ->
//       mix=M(I-M+eps)^-1 -> B[k][j] = T2[j][k] = -kappa[k]*(0.5*d_jk+mix[j][k])
//   Phase 2 (integrate_kernel, 64 blocks x 128 thr): fixed-step RK4; batch rows
//       decouple; complex GEMM (16x64)@(64x64) per stage via v_wmma_f32_16x16x4_f32.
//       B fragments live in registers for the whole integration (invariant),
//       so the inner loop touches LDS only for the A (state) fragments.
// ---------------------------------------------------------------------------

#define N_MODES   64
#define N_IN      48
#define N_BATCH   1024
#define N_EVAL    200
#define ROWS_PB   16
#define ITHREADS  128
#define STHREADS  256
#define NSUB      16
#define EXPM_S    12
#define EXPM_K    18

typedef float v2f __attribute__((ext_vector_type(2)));
typedef float v8f __attribute__((ext_vector_type(8)));

// workspace layout (float offsets)
#define WS_HRE   0
#define WS_HIM   4096
#define WS_XRE   8192
#define WS_XIM   12288
#define WS_RRE   16384
#define WS_RIM   20480
#define WS_TRE   24576
#define WS_TIM   28672
#define WS_MRE   32768
#define WS_MIM   36864
#define WS_AGRE  40960   // 64x128
#define WS_AGIM  49152   // 64x128
#define WS_FACR  57344   // 64
#define WS_FACI  57408   // 64
#define WS_BRE   57472   // 64x64
#define WS_BIM   61568   // 64x64
// total: 65664 floats (~257 KB) of d_ws

static __device__ __forceinline__ v8f wmma_f32_k4(v2f a, v2f b, v8f c) {
    // 8-arg form: (neg_a, A, neg_b, B, c_mod, C, reuse_a, reuse_b)
    return __builtin_amdgcn_wmma_f32_16x16x4_f32(
        false, a, false, b, (short)0, c, false, false);
}

// 64x64 complex matmul, one workgroup, results to Cr/Ci (global scratch)
static __device__ __forceinline__ void cgemm64(
    const float* __restrict__ Ar, const float* __restrict__ Ai,
    const float* __restrict__ Br, const float* __restrict__ Bi,
    float* __restrict__ Cr, float* __restrict__ Ci, int t)
{
    for (int e = t; e < 4096; e += STHREADS) {
        int r = e >> 6, c = e & 63;
        float sr = 0.f, si = 0.f;
        for (int k = 0; k < 64; ++k) {
            float ar = Ar[r*64+k], ai = Ai[r*64+k];
            float br = Br[k*64+c], bi = Bi[k*64+c];
            sr = fmaf(ar, br, fmaf(-ai, bi, sr));
            si = fmaf(ar, bi, fmaf( ai, br, si));
        }
        Cr[e] = sr; Ci[e] = si;
    }
}

__global__ void __launch_bounds__(STHREADS)
setup_kernel(const float* __restrict__ params,
             const float* __restrict__ kappa,
             float* __restrict__ ws)
{
    const int t = threadIdx.x;
    float* Hre = ws + WS_HRE;  float* Him = ws + WS_HIM;
    float* Xre = ws + WS_XRE;  float* Xim = ws + WS_XIM;
    float* Rre = ws + WS_RRE;  float* Rim = ws + WS_RIM;
    float* Tre = ws + WS_TRE;  float* Tim = ws + WS_TIM;
    float* Mre = ws + WS_MRE;  float* Mim = ws + WS_MIM;
    float* AgR = ws + WS_AGRE; float* AgI = ws + WS_AGIM;
    float* FcR = ws + WS_FACR; float* FcI = ws + WS_FACI;
    float* Bre = ws + WS_BRE;  float* Bim = ws + WS_BIM;

    // ---- build Hermitian H from su(64) coefficients ----
    for (int e = t; e < 4096; e += STHREADS) { Hre[e] = 0.f; Him[e] = 0.f; }
    __syncthreads();
    for (int idx = t; idx < 2016; idx += STHREADS) {       // triu(i<j), row-major
        int i = 0, rem = idx;
        while (rem >= 63 - i) { rem -= 63 - i; ++i; }
        int j = i + 1 + rem;
        float re = params[idx], im = params[2016 + idx];
        Hre[i*64+j] =  re; Him[i*64+j] =  im;
        Hre[j*64+i] =  re; Him[j*64+i] = -im;
    }
    __syncthreads();
    if (t < 64) {
        float d;
        if (t < 63) d = params[4032 + t];
        else { float s = 0.f; for (int k = 0; k < 63; ++k) s += params[4032+k]; d = -s; }
        Hre[t*64+t] += d;
    }
    __syncthreads();

    // ---- U = expm(iH): scaling & squaring + Taylor/Horner ----
    const float sc = 1.0f / (float)(1 << EXPM_S);
    for (int e = t; e < 4096; e += STHREADS) {             // X = iH / 2^s
        Xre[e] = -Him[e] * sc;  Xim[e] = Hre[e] * sc;
    }
    for (int e = t; e < 4096; e += STHREADS) {             // R = I
        Rre[e] = ((e >> 6) == (e & 63)) ? 1.f : 0.f;  Rim[e] = 0.f;
    }
    __syncthreads();
    for (int k = EXPM_K; k >= 1; --k) {                    // R = I + X*R/k
        cgemm64(Xre, Xim, Rre, Rim, Tre, Tim, t);
        __syncthreads();
        const float ik = 1.0f / (float)k;
        for (int e = t; e < 4096; e += STHREADS) {
            float dr = Tre[e] * ik, di = Tim[e] * ik;
            if ((e >> 6) == (e & 63)) dr += 1.f;
            Rre[e] = dr; Rim[e] = di;
        }
        __syncthreads();
    }
    for (int sq = 0; sq < EXPM_S; ++sq) {                  // R = R*R  (s times)
        cgemm64(Rre, Rim, Rre, Rim, Tre, Tim, t);
        __syncthreads();
        for (int e = t; e < 4096; e += STHREADS) { Rre[e] = Tre[e]; Rim[e] = Tim[e]; }
        __syncthreads();
    }

    // ---- M = U^T U  (plain transpose, no conjugate) ----
    for (int e = t; e < 4096; e += STHREADS) {
        int r = e >> 6, c = e & 63;
        float sr = 0.f, si = 0.f;
        for (int k = 0; k < 64; ++k) {
            float ar = Rre[k*64+r], ai = Rim[k*64+r];
            float br = Rre[k*64+c], bi = Rim[k*64+c];
            sr = fmaf(ar, br, fmaf(-ai, bi, sr));
            si = fmaf(ar, bi, fmaf( ai, br, si));
        }
        Mre[e] = sr; Mim[e] = si;
    }
    __syncthreads();

    // ---- Gauss-Jordan inverse of C = (1+eps)I - M, augmented [C | I] ----
    for (int e = t; e < 8192; e += STHREADS) {
        int r = e >> 7, c = e & 127;
        float vr, vi;
        if (c < 64) { vr = ((r == c) ? (1.f + 1e-8f) : 0.f) - Mre[r*64+c]; vi = -Mim[r*64+c]; }
        else        { vr = ((c - 64) == r) ? 1.f : 0.f;                     vi = 0.f; }
        AgR[e] = vr; AgI[e] = vi;
    }
    __syncthreads();
    for (int p = 0; p < 64; ++p) {
        float pr = AgR[p*128+p], pi = AgI[p*128+p];
        float den = pr*pr + pi*pi + 1e-30f;
        float ivr = pr / den, ivi = -pi / den;
        __syncthreads();
        if (t < 128) {                                      // normalize pivot row
            float vr = AgR[p*128+t], vi = AgI[p*128+t];
            AgR[p*128+t] = vr*ivr - vi*ivi;
            AgI[p*128+t] = vr*ivi + vi*ivr;
        }
        __syncthreads();
        if (t < 64) { FcR[t] = AgR[t*128+p]; FcI[t] = AgI[t*128+p]; }
        __syncthreads();
        for (int e = t; e < 8192; e += STHREADS) {          // eliminate
            int r = e >> 7, c = e & 127;
            if (r != p) {
                float fr = FcR[r], fi = FcI[r];
                float qr = AgR[p*128+c], qi = AgI[p*128+c];
                AgR[e] -= fr*qr - fi*qi;
                AgI[e] -= fr*qi + fi*qr;
            }
        }
        __syncthreads();
    }

    // ---- mix = M @ Cinv (reuse X buffers) ----
    for (int e = t; e < 4096; e += STHREADS) {
        int r = e >> 6, c = e & 63;
        float sr = 0.f, si = 0.f;
        for (int k = 0; k < 64; ++k) {
            float ar = Mre[r*64+k],       ai = Mim[r*64+k];
            float br = AgR[k*128+64+c],   bi = AgI[k*128+64+c];
            sr = fmaf(ar, br, fmaf(-ai, bi, sr));
            si = fmaf(ar, bi, fmaf( ai, br, si));
        }
        Xre[e] = sr; Xim[e] = si;
    }
    __syncthreads();

    // ---- GEMM B-matrix: B[k][j] = T2[j][k] = -kappa[k]*(0.5*(j==k)+mix[j][k]) ----
    for (int e = t; e < 4096; e += STHREADS) {
        int k = e >> 6, j = e & 63;
        float mr = Xre[j*64+k] + ((j == k) ? 0.5f : 0.f);
        float mi = Xim[j*64+k];
        float kk = kappa[k];
        Bre[e] = -kk * mr;
        Bim[e] = -kk * mi;
    }
}

// ---------------------------------------------------------------------------
// Integration kernel: block = 16 batch rows x 64 modes; 4 waves, each wave
// owns one 16x16 WMMA tile of the complex GEMM (K=64 -> 16 x v_wmma 16x16x4).
// B fragments are loaded once into registers; LDS is only Y/D staging (16 KB).
// ---------------------------------------------------------------------------
__global__ void __launch_bounds__(ITHREADS)
integrate_kernel(const float* __restrict__ A0,
                 const float* __restrict__ omega,
                 const float* __restrict__ nonlin,
                 const float* __restrict__ ws,
                 float* __restrict__ out)
{
    __shared__ float Yre[ROWS_PB*64], Yim[ROWS_PB*64];   // stage input  (8 KB)
    __shared__ float Dre[ROWS_PB*64], Dim[ROWS_PB*64];   // GEMM result  (8 KB)

    const int t    = threadIdx.x;
    const int b0   = blockIdx.x * ROWS_PB;
    const int r    = t >> 3;            // 0..15  local batch row
    const int c0   = (t & 7) * 8;       // owned mode columns [c0, c0+8)
    const int lane = t & 31;
    const int wave = t >> 5;            // 0..3
    const int C0   = wave * 16;         // wave's N tile
    const int lm   = lane & 15;         // M (A frag) / N (B,C frag)
    const int half = lane >> 4;         // K split within frag

    // B (T2^T) fragments: invariant across the whole integration -> registers.
    const float* wsBre = ws + WS_BRE;
    const float* wsBim = ws + WS_BIM;
    v2f bR[16], bI[16];
#pragma unroll
    for (int kc = 0; kc < 16; ++kc) {
        const int ka = kc*4 + half*2;
        bR[kc].x = wsBre[ ka   *64 + C0 + lm];
        bR[kc].y = wsBre[(ka+1)*64 + C0 + lm];
        bI[kc].x = wsBim[ ka   *64 + C0 + lm];
        bI[kc].y = wsBim[(ka+1)*64 + C0 + lm];
    }

    // per-thread frequency constants
    float om[8], nl[8];
#pragma unroll
    for (int j = 0; j < 8; ++j) { om[j] = omega[c0 + j]; nl[j] = nonlin[c0 + j]; }

    // initial condition
    const int b = b0 + r;
    float yR[8], yI[8];
#pragma unroll
    for (int j = 0; j < 8; ++j) {
        int c = c0 + j;
        if (c < N_IN) { yR[j] = A0[(b*N_IN + c)*2 + 0]; yI[j] = A0[(b*N_IN + c)*2 + 1]; }
        else          { yR[j] = 1.f;                    yI[j] = 0.f; }
    }

    // one RHS evaluation: in -> out (registers), complex GEMM via WMMA
    auto stage = [&](const float* inR, const float* inI, float* oR, float* oI) {
        // publish stage input (A matrix, row-major in LDS)
#pragma unroll
        for (int j = 0; j < 8; ++j) {
            Yre[r*64 + c0 + j] = inR[j];
            Yim[r*64 + c0 + j] = inI[j];
        }
        __syncthreads();

        // complex 16x64 @ 64x16 tile: 4 real accumulator chains
        v8f cRR = {}, cII = {}, cRI = {}, cIR = {};
#pragma unroll
        for (int kc = 0; kc < 16; ++kc) {
            const int ka = kc*4 + half*2;
            v2f ar, ai;
            // A frag: lane lm = row M, VGPR0/1 = K = ka, ka+1
            ar.x = Yre[lm*64 + ka];  ar.y = Yre[lm*64 + ka + 1];
            ai.x = Yim[lm*64 + ka];  ai.y = Yim[lm*64 + ka + 1];
            cRR = wmma_f32_k4(ar, bR[kc], cRR);
            cII = wmma_f32_k4(ai, bI[kc], cII);
            cRI = wmma_f32_k4(ar, bI[kc], cRI);
            cIR = wmma_f32_k4(ai, bR[kc], cIR);
        }
        // store D tile per C/D layout: VGPR v -> row v + half*8, col lm
#pragma unroll
        for (int v = 0; v < 8; ++v) {
            Dre[(v + half*8)*64 + C0 + lm] = cRR[v] - cII[v];
            Dim[(v + half*8)*64 + C0 + lm] = cRI[v] + cIR[v];
        }
        __syncthreads();

        // gather + nonlinear frequency term: d = i(w + g|A|^2)A + A@T2^T
#pragma unroll
        for (int j = 0; j < 8; ++j) {
            float gr = Dre[r*64 + c0 + j];
            float gi = Dim[r*64 + c0 + j];
            float G  = om[j] + nl[j] * (inR[j]*inR[j] + inI[j]*inI[j]);
            oR[j] = gr - G * inI[j];
            oI[j] = gi + G * inR[j];
        }
    };

    const float dt = 2.0f / (float)(N_EVAL - 1);
    const float h  = dt / (float)NSUB;

    for (int ev = 0; ev < N_EVAL; ++ev) {
        // record state at this eval point (16 consecutive floats -> 4x b128)
        float4* o4 = (float4*)(out + (((size_t)ev * N_BATCH + b) * N_MODES + c0) * 2);
#pragma unroll
        for (int j = 0; j < 4; ++j) {
            float4 v; v.x = yR[2*j]; v.y = yI[2*j]; v.z = yR[2*j+1]; v.w = yI[2*j+1];
            o4[j] = v;
        }
        if (ev == N_EVAL - 1) break;

        for (int s = 0; s < NSUB; ++s) {
            float aR[8], aI[8], sR[8], sI[8], dR[8], dI[8];
            // k1
            stage(yR, yI, dR, dI);
#pragma unroll
            for (int j = 0; j < 8; ++j) {
                aR[j] = dR[j];                aI[j] = dI[j];
                sR[j] = yR[j] + 0.5f*h*dR[j]; sI[j] = yI[j] + 0.5f*h*dI[j];
            }
            // k2
            stage(sR, sI, dR, dI);
#pragma unroll
            for (int j = 0; j < 8; ++j) {
                aR[j] += 2.f*dR[j];           aI[j] += 2.f*dI[j];
                sR[j] = yR[j] + 0.5f*h*dR[j]; sI[j] = yI[j] + 0.5f*h*dI[j];
            }
            // k3
            stage(sR, sI, dR, dI);
#pragma unroll
            for (int j = 0; j < 8; ++j) {
                aR[j] += 2.f*dR[j];           aI[j] += 2.f*dI[j];
                sR[j] = yR[j] + h*dR[j];      sI[j] = yI[j] + h*dI[j];
            }
            // k4 + combine
            stage(sR, sI, dR, dI);
            const float h6 = h / 6.f;
#pragma unroll
            for (int j = 0; j < 8; ++j) {
                yR[j] += h6 * (aR[j] + dR[j]);
                yI[j] += h6 * (aI[j] + dI[j]);
            }
        }
    }
}

extern "C" void kernel_launch(void* const* d_in, const int* in_sizes, int n_in,
                              void* d_out, int out_size, void* d_ws, size_t ws_size,
                              hipStream_t stream) {
    const float* A0     = (const float*)d_in[0];   // (1024, 48, 2)
    const float* params = (const float*)d_in[1];   // (4095,)
    const float* omega  = (const float*)d_in[2];   // (64,)
    const float* kappa  = (const float*)d_in[3];   // (64,)
    const float* nonlin = (const float*)d_in[4];   // (64,)
    float*       out    = (float*)d_out;           // (200, 1024, 64, 2)
    float*       ws     = (float*)d_ws;

    setup_kernel<<<1, STHREADS, 0, stream>>>(params, kappa, ws);
    integrate_kernel<<<N_BATCH / ROWS_PB, ITHREADS, 0, stream>>>(
        A0, omega, nonlin, ws, out);
}